// FGL_82480551952944
// MI455X (gfx1250) — compile-verified
//
#include <hip/hip_runtime.h>

typedef float v2f __attribute__((ext_vector_type(2)));
typedef float v8f __attribute__((ext_vector_type(8)));

#define N_    32
#define INC_  32
#define INN_  16384
#define OUTC_ 32
#define OUTN_ 4096
#define MAXD_ 16
#define OT_   (OUTN_ / 16)   // 256 node-tiles per batch element

// One wave32 per block. Each block handles batch n, a tile of 16 output
// nodes, and all 32 output channels.
//   Phase 1: gather+pool directly into WMMA A-fragment layout (f32, 16x4 tiles)
//   Phase 2: chain 8x V_WMMA_F32_16X16X4_F32 per 16-wide c-half (K = INC = 32),
//            with C initialized from bias (free bias add), then store D.
__global__ __launch_bounds__(32)
void FGL_fused_gather_wmma(const float* __restrict__ x,
                           const int*   __restrict__ A,
                           const float* __restrict__ mask,
                           const float* __restrict__ w,
                           const float* __restrict__ bias,
                           float* __restrict__ out)
{
    const int tile = blockIdx.x;
    const int n    = tile / OT_;
    const int ob   = (tile % OT_) * 16;

    const int lane = threadIdx.x;   // 0..31, wave32
    const int mlo  = lane & 15;     // M (node row) for A; N (channel col) for B/C/D
    const int hi   = lane >> 4;     // selects K sub-pair per ISA fragment layout

    const int o = ob + mlo;

    // Per-lane copy of this node's adjacency row and mask row.
    int   idx[MAXD_];
    float msk[MAXD_];
#pragma unroll
    for (int m = 0; m < MAXD_; ++m) {
        idx[m] = A[o * MAXD_ + m];
        msk[m] = mask[o * MAXD_ + m];
    }

    // A-fragment for V_WMMA_F32_16X16X4_F32 step s:
    //   lane holds M = mlo, VGPR0 -> K = 4s + 2*hi, VGPR1 -> K = 4s + 2*hi + 1
    // where K indexes the input channel i. pooled[o][i] is computed in place.
    v2f afrag[8];
    const size_t xbase = (size_t)n * INC_ * INN_;
#pragma unroll
    for (int s = 0; s < 8; ++s) {
        const int i0 = 4 * s + 2 * hi;
        const float* __restrict__ row0 = x + xbase + (size_t)i0 * INN_;
        const float* __restrict__ row1 = row0 + INN_;
        float acc0 = 0.0f, acc1 = 0.0f;
#pragma unroll
        for (int m = 0; m < MAXD_; ++m) {
            acc0 = fmaf(row0[idx[m]], msk[m], acc0);
            acc1 = fmaf(row1[idx[m]], msk[m], acc1);
        }
        afrag[s].x = acc0;
        afrag[s].y = acc1;
    }

    // Two 16-wide halves of the 32 output channels.
#pragma unroll
    for (int h = 0; h < 2; ++h) {
        const int c = h * 16 + mlo;   // this lane's N (channel) in B/C/D

        // C init from bias: C VGPR v holds (M = v + 8*hi, N = c)
        v8f acc;
#pragma unroll
        for (int v = 0; v < 8; ++v)
            acc[v] = bias[(size_t)c * OUTN_ + (ob + v + 8 * hi)];

        // K-loop: 8 steps of K=4 cover INC=32.
#pragma unroll
        for (int s = 0; s < 8; ++s) {
            const int k0 = 4 * s + 2 * hi;
            v2f bfrag;
            bfrag.x = w[(size_t)k0       * OUTC_ + c];
            bfrag.y = w[(size_t)(k0 + 1) * OUTC_ + c];
            acc = __builtin_amdgcn_wmma_f32_16x16x4_f32(
                /*neg_a=*/false, afrag[s],
                /*neg_b=*/false, bfrag,
                /*c_mod=*/(short)0, acc,
                /*reuse_a=*/false, /*reuse_b=*/false);
        }

        // Store D: VGPR v -> out[n, c, ob + v + 8*hi]; 8 contiguous floats
        // per lane -> merges into wide global stores.
        float* __restrict__ op =
            out + ((size_t)n * OUTC_ + c) * OUTN_ + ob + 8 * hi;
#pragma unroll
        for (int v = 0; v < 8; ++v)
            op[v] = acc[v];
    }
}

extern "C" void kernel_launch(void* const* d_in, const int* in_sizes, int n_in,
                              void* d_out, int out_size, void* d_ws, size_t ws_size,
                              hipStream_t stream) {
    const float* x    = (const float*)d_in[0];
    const int*   A    = (const int*)  d_in[1];
    const float* mask = (const float*)d_in[2];
    const float* w    = (const float*)d_in[3];
    const float* bias = (const float*)d_in[4];
    float* out = (float*)d_out;

    dim3 grid(N_ * OT_);   // 8192 blocks
    dim3 block(32);        // one wave32 each
    hipLaunchKernelGGL(FGL_fused_gather_wmma, grid, block, 0, stream,
                       x, A, mask, w, bias, out);
}